// HybridGATGCNWithIGFT_9371618639990
// MI455X (gfx1250) — compile-verified
//
#include <hip/hip_runtime.h>
#include <math.h>

#define NN      4096
#define F_INP   256
#define GCN_H   512
#define GAT_H   256
#define HEADS   4
#define HID     (HEADS*GAT_H)   // 1024
#define OUT_DIM 64
#define NG      32
#define NEG_SLOPE 0.2f

// ---- block one-sided Jacobi parameters ----
#define NB      64     // number of 64-wide column panels
#define PW      128    // columns per pair (2 panels)
#define NPAIR   32     // pairs per round
#define NROUND  63     // rounds per outer sweep (NB-1)
#define OUTER_SWEEPS 10
#define INNER_SWEEPS 5

#define GS  132        // padded stride for 128-wide Jacobi LDS matrices
#define TS2 36         // Gram staging tile: [col][k] 128 x (32+4)
#define JS  132        // J in apply_rot: [n][k] 128 x (128+4)
#define CS  132        // C tile in apply_rot: [m][k] 64 x (128+4)

typedef __attribute__((ext_vector_type(2))) float v2f;
typedef __attribute__((ext_vector_type(8))) float v8f;

__device__ __forceinline__ v8f wmma_f32(v2f a, v2f b, v8f c) {
  // D = A(16x4,f32) * B(4x16,f32) + C(16x16,f32)
  return __builtin_amdgcn_wmma_f32_16x16x4_f32(false, a, false, b, (short)0, c,
                                               false, false);
}

// round-robin tournament pairing (circle method)
__device__ __forceinline__ void outer_pair(int r, int i, int* p, int* q) {
  int a = (r + i) % (NB - 1);
  int b = (i == 0) ? (NB - 1) : (r - i + 2 * (NB - 1)) % (NB - 1);
  *p = a; *q = b;
}
__device__ __forceinline__ void inner_pair(int r, int i, int* p, int* q) {
  int a = (r + i) % (PW - 1);
  int b = (i == 0) ? (PW - 1) : (r - i + 2 * (PW - 1)) % (PW - 1);
  *p = a; *q = b;
}

// ---------------- utility kernels ----------------
__global__ void k_fill(float* p, float v, long n) {
  long i = (long)blockIdx.x * blockDim.x + threadIdx.x;
  if (i < n) p[i] = v;
}

__global__ void k_set_adj(float* A, const int* s, const int* d, int E) {
  int e = blockIdx.x * blockDim.x + threadIdx.x;
  if (e < E) A[(long)s[e] * NN + d[e]] = 1.0f;
}

__global__ void k_row_sum(const float* __restrict__ A, float* deg) {
  __shared__ float red[256];
  int row = blockIdx.x;
  float s = 0.f;
  for (int j = threadIdx.x; j < NN; j += 256) s += A[(long)row * NN + j];
  red[threadIdx.x] = s; __syncthreads();
  for (int k = 128; k > 0; k >>= 1) {
    if (threadIdx.x < k) red[threadIdx.x] += red[threadIdx.x + k];
    __syncthreads();
  }
  if (threadIdx.x == 0) deg[row] = red[0];
}

__global__ void k_form_lap(float* A, const float* deg) {
  long i = (long)blockIdx.x * blockDim.x + threadIdx.x;
  if (i < (long)NN * NN) {
    int r = (int)(i / NN), c = (int)(i % NN);
    float a = A[i];
    A[i] = ((r == c) ? deg[r] : 0.f) - a;
  }
}

__global__ void k_set_diag(float* V) {
  int i = blockIdx.x * blockDim.x + threadIdx.x;
  if (i < NN) V[(long)i * NN + i] = 1.f;
}

// ---------------- K1: Gram (WMMA) + inner Jacobi on 128x128 ----------------
__global__ void k_gram_jacobi(const float* __restrict__ A, float* __restrict__ Jws,
                              int round) {
  extern __shared__ float smem[];
  float* Gl = smem;                 // 128*GS   [row][col]
  float* Jl = Gl + 128 * GS;        // 128*GS   [row][col]
  float* T  = Jl + 128 * GS;        // 128*TS2  [col][k]  (fragment-major)
  float* cc = T + 128 * TS2;        // 64
  float* ss = cc + 64;              // 64
  int*   pp = (int*)(ss + 64);      // 64
  int*   qq = pp + 64;              // 64

  int tid = threadIdx.x;
  int wv = tid >> 5, ln = tid & 31;
  int l15 = ln & 15, lh = ln >> 4;
  int pair = blockIdx.x;
  int bp, bq; outer_pair(round, pair, &bp, &bq);

  v8f acc[8];
  v8f zero = {0.f,0.f,0.f,0.f,0.f,0.f,0.f,0.f};
#pragma unroll
  for (int j = 0; j < 8; ++j) acc[j] = zero;

  // ---- Gram G = C^T C over the 128 selected columns, WMMA f32 ----
  for (int kb = 0; kb < NN; kb += 32) {
    // stage T[col][k]: 4 k-strided coalesced global loads -> one b128 LDS store
    for (int idx = tid; idx < 128 * 8; idx += 256) {
      int col = idx & 127;
      int k4  = (idx >> 7) << 2;
      int blk = (col < 64) ? bp : bq;
      int co  = (col < 64) ? col : col - 64;
      const float* g = A + (long)(kb + k4) * NN + blk * 64 + co;
      float4 v;
      v.x = g[0]; v.y = g[NN]; v.z = g[2 * NN]; v.w = g[3 * NN];
      *(float4*)&T[col * TS2 + k4] = v;
    }
    __syncthreads();
#pragma unroll
    for (int k0 = 0; k0 < 32; k0 += 4) {
      int kk2 = k0 + (lh << 1);
      v2f fa = *(const v2f*)&T[(wv * 16 + l15) * TS2 + kk2];
#pragma unroll
      for (int j = 0; j < 8; ++j) {
        v2f fb = *(const v2f*)&T[(j * 16 + l15) * TS2 + kk2];
        acc[j] = wmma_f32(fa, fb, acc[j]);
      }
    }
    __syncthreads();
  }
  // spill G to LDS, init J = I
#pragma unroll
  for (int j = 0; j < 8; ++j)
#pragma unroll
    for (int r = 0; r < 8; ++r)
      Gl[(wv * 16 + r + (lh << 3)) * GS + j * 16 + l15] = acc[j][r];
  for (int idx = tid; idx < 128 * 128; idx += 256) {
    int a = idx >> 7, b = idx & 127;
    Jl[a * GS + b] = (a == b) ? 1.f : 0.f;
  }
  __syncthreads();

  // ---- two-sided Jacobi on G, accumulate rotations into J ----
  for (int sw = 0; sw < INNER_SWEEPS; ++sw) {
    for (int r2 = 0; r2 < PW - 1; ++r2) {
      if (tid < 64) {
        int p, q; inner_pair(r2, tid, &p, &q);
        pp[tid] = p; qq[tid] = q;
        float app = Gl[p * GS + p], aqq = Gl[q * GS + q], apq = Gl[p * GS + q];
        float c = 1.f, s = 0.f;
        if (fabsf(apq) > 1e-20f) {
          float tau = (aqq - app) / (2.f * apq);
          float t = copysignf(1.f, tau) / (fabsf(tau) + sqrtf(1.f + tau * tau));
          c = rsqrtf(1.f + t * t);
          s = t * c;
        }
        cc[tid] = c; ss[tid] = s;
      }
      __syncthreads();
      // column pass: G <- G R ; J <- J R
      for (int idx = tid; idx < 128 * 64; idx += 256) {
        int pr = idx & 63, row = idx >> 6;
        int p = pp[pr], q = qq[pr];
        float c = cc[pr], s = ss[pr];
        float gp = Gl[row * GS + p], gq = Gl[row * GS + q];
        Gl[row * GS + p] = c * gp - s * gq;
        Gl[row * GS + q] = s * gp + c * gq;
        float jp = Jl[row * GS + p], jq = Jl[row * GS + q];
        Jl[row * GS + p] = c * jp - s * jq;
        Jl[row * GS + q] = s * jp + c * jq;
      }
      __syncthreads();
      // row pass: G <- R^T G
      for (int idx = tid; idx < 128 * 64; idx += 256) {
        int pr = idx & 63, col = idx >> 6;
        int p = pp[pr], q = qq[pr];
        float c = cc[pr], s = ss[pr];
        float gp = Gl[p * GS + col], gq = Gl[q * GS + col];
        Gl[p * GS + col] = c * gp - s * gq;
        Gl[q * GS + col] = s * gp + c * gq;
      }
      __syncthreads();
    }
  }
  float* Jo = Jws + (long)pair * PW * PW;
  for (int idx = tid; idx < 128 * 128; idx += 256)
    Jo[idx] = Jl[(idx >> 7) * GS + (idx & 127)];
}

// ---------------- K2: panel update C <- C*J, V <- V*J (WMMA) ----------------
__global__ void k_apply_rot(float* __restrict__ A, float* __restrict__ V,
                            const float* __restrict__ Jws, int round) {
  extern __shared__ float smem[];
  float* Jl = smem;              // 128*JS  [n][k]  (fragment-major)
  float* Ct = Jl + 128 * JS;     // 64*CS   [m][k]  (fragment-major)
  int tid = threadIdx.x, wv = tid >> 5, ln = tid & 31;
  int l15 = ln & 15, lh = ln >> 4;
  int pair = blockIdx.x, rt = blockIdx.y;
  int bp, bq; outer_pair(round, pair, &bp, &bq);
  const float* Jo = Jws + (long)pair * PW * PW;   // row-major [k][n]
  for (int idx = tid; idx < 128 * 32; idx += 256) {
    int n  = idx & 127;
    int k4 = (idx >> 7) << 2;
    const float* g = Jo + (long)k4 * 128 + n;
    float4 v;
    v.x = g[0]; v.y = g[128]; v.z = g[256]; v.w = g[384];
    *(float4*)&Jl[n * JS + k4] = v;
  }
  float* mats[2]; mats[0] = A; mats[1] = V;
  int tr = wv & 3, tcb = (wv >> 2) << 2;  // wave -> (row tile, 4 col tiles)
  v8f zero = {0.f,0.f,0.f,0.f,0.f,0.f,0.f,0.f};
  for (int m2 = 0; m2 < 2; ++m2) {
    float* M = mats[m2];
    __syncthreads();
    for (int idx = tid; idx < 64 * 32; idx += 256) {
      int mr = idx >> 5; int c4 = (idx & 31) << 2;
      int blk = (c4 < 64) ? bp : bq;
      int co  = (c4 < 64) ? c4 : c4 - 64;
      float4 v = *(const float4*)(M + (long)(rt * 64 + mr) * NN + blk * 64 + co);
      *(float4*)&Ct[mr * CS + c4] = v;
    }
    __syncthreads();
    v8f acc[4];
#pragma unroll
    for (int j = 0; j < 4; ++j) acc[j] = zero;
    for (int k0 = 0; k0 < 128; k0 += 4) {
      int kk2 = k0 + (lh << 1);
      v2f a = *(const v2f*)&Ct[(tr * 16 + l15) * CS + kk2];
#pragma unroll
      for (int j = 0; j < 4; ++j) {
        v2f b = *(const v2f*)&Jl[((tcb + j) * 16 + l15) * JS + kk2];
        acc[j] = wmma_f32(a, b, acc[j]);
      }
    }
#pragma unroll
    for (int j = 0; j < 4; ++j) {
      int ci = (tcb + j) * 16 + l15;
      int gcol = (ci < 64) ? bp * 64 + ci : bq * 64 + ci - 64;
#pragma unroll
      for (int r = 0; r < 8; ++r) {
        int grow = rt * 64 + tr * 16 + r + (lh << 3);
        M[(long)grow * NN + gcol] = acc[j][r];
      }
    }
  }
}

// ---------------- eigenvalue extraction / sort / permute ----------------
__global__ void k_col_norm(const float* __restrict__ A, float* lam) {
  __shared__ float red[256];
  int col = blockIdx.x;
  float s = 0.f;
  for (int r = threadIdx.x; r < NN; r += 256) {
    float v = A[(long)r * NN + col]; s += v * v;
  }
  red[threadIdx.x] = s; __syncthreads();
  for (int k = 128; k > 0; k >>= 1) {
    if (threadIdx.x < k) red[threadIdx.x] += red[threadIdx.x + k];
    __syncthreads();
  }
  if (threadIdx.x == 0) lam[col] = sqrtf(red[0]);
}

__global__ void k_iota(int* idx) {
  int i = blockIdx.x * blockDim.x + threadIdx.x;
  if (i < NN) idx[i] = i;
}

__global__ void k_bitonic(float* key, int* idx, int j, int k) {
  int i = blockIdx.x * blockDim.x + threadIdx.x;
  int ixj = i ^ j;
  if (i < NN && ixj > i) {
    bool asc = ((i & k) == 0);
    float ki = key[i], kj = key[ixj];
    bool sw = asc ? (ki > kj) : (ki < kj);
    if (sw) {
      key[i] = kj; key[ixj] = ki;
      int t = idx[i]; idx[i] = idx[ixj]; idx[ixj] = t;
    }
  }
}

__global__ void k_permute_cols(const float* __restrict__ V, float* __restrict__ U,
                               const int* __restrict__ idx) {
  long i = (long)blockIdx.x * blockDim.x + threadIdx.x;
  if (i < (long)NN * NN) {
    int r = (int)(i / NN), c = (int)(i % NN);
    U[i] = V[(long)r * NN + idx[c]];
  }
}

// ---------------- generic f32 WMMA GEMM: C = op(A)*B ----------------
// requires M%64==0, N%64==0, K%16==0
__global__ void k_gemm(const float* __restrict__ A, const float* __restrict__ B,
                       float* __restrict__ C, int M, int Ncols, int K,
                       int lda, int transA) {
  // fragment-major LDS tiles: As[m][k], Bs[n][k], stride 20 (16+4 pad)
  __shared__ float As[64][20];
  __shared__ float Bs[64][20];
  int tid = threadIdx.x, wv = tid >> 5, ln = tid & 31;
  int l15 = ln & 15, lh = ln >> 4;
  int bm = blockIdx.y * 64, bn = blockIdx.x * 64;
  int tr = wv & 3, tcb = (wv >> 2) << 1;  // two 16x16 tiles per wave
  v8f acc0 = {0.f,0.f,0.f,0.f,0.f,0.f,0.f,0.f};
  v8f acc1 = {0.f,0.f,0.f,0.f,0.f,0.f,0.f,0.f};
  int mA = tid >> 2, k4 = (tid & 3) << 2;
  for (int kk = 0; kk < K; kk += 16) {
    if (!transA) {
      float4 v = *(const float4*)(A + (long)(bm + mA) * lda + kk + k4);
      *(float4*)&As[mA][k4] = v;
      if (kk + 16 < K)
        __builtin_prefetch(A + (long)(bm + mA) * lda + kk + 16 + k4, 0, 1);
    } else {
      const float* g = A + (long)(kk + k4) * lda + bm + mA;
      float4 v;
      v.x = g[0]; v.y = g[lda]; v.z = g[2 * (long)lda]; v.w = g[3 * (long)lda];
      *(float4*)&As[mA][k4] = v;
      if (kk + 16 < K)
        __builtin_prefetch(A + (long)(kk + 16 + k4) * lda + bm + mA, 0, 1);
    }
    {
      const float* g = B + (long)(kk + k4) * Ncols + bn + mA;
      float4 v;
      v.x = g[0]; v.y = g[Ncols]; v.z = g[2 * (long)Ncols]; v.w = g[3 * (long)Ncols];
      *(float4*)&Bs[mA][k4] = v;
      if (kk + 16 < K)
        __builtin_prefetch(B + (long)(kk + 16 + k4) * Ncols + bn + mA, 0, 1);
    }
    __syncthreads();
#pragma unroll
    for (int k0 = 0; k0 < 16; k0 += 4) {
      int kk2 = k0 + (lh << 1);
      v2f a  = *(const v2f*)&As[tr * 16 + l15][kk2];
      v2f b0 = *(const v2f*)&Bs[tcb * 16 + l15][kk2];
      v2f b1 = *(const v2f*)&Bs[(tcb + 1) * 16 + l15][kk2];
      acc0 = wmma_f32(a, b0, acc0);
      acc1 = wmma_f32(a, b1, acc1);
    }
    __syncthreads();
  }
#pragma unroll
  for (int r = 0; r < 8; ++r) {
    int row = bm + tr * 16 + r + (lh << 3);
    C[(long)row * Ncols + bn + tcb * 16 + l15] = acc0[r];
    C[(long)row * Ncols + bn + (tcb + 1) * 16 + l15] = acc1[r];
  }
}

// ---------------- GCN / GAT sparse parts ----------------
__global__ void k_bias_relu(float* p, const float* b, long n, int cols) {
  long i = (long)blockIdx.x * blockDim.x + threadIdx.x;
  if (i < n) {
    float v = p[i] + b[i % cols];
    p[i] = v > 0.f ? v : 0.f;
  }
}

__global__ void k_edge_deg(const int* d, int E, float* deg2) {
  int e = blockIdx.x * blockDim.x + threadIdx.x;
  if (e < E) atomicAdd(&deg2[d[e]], 1.0f);
}

__global__ void k_rsqrt(float* p, int n) {
  int i = blockIdx.x * blockDim.x + threadIdx.x;
  if (i < n) p[i] = rsqrtf(p[i]);
}

__global__ void k_gcn_scatter(const float* __restrict__ hw, const int* s,
                              const int* d, const float* dis, float* out, int E) {
  int e = blockIdx.x;
  if (e >= E) return;
  int si = s[e], di = d[e];
  float w = dis[si] * dis[di];
  const float* src = hw + (long)si * GCN_H;
  float* dst = out + (long)di * GCN_H;
  for (int f = threadIdx.x; f < GCN_H; f += blockDim.x)
    atomicAdd(&dst[f], src[f] * w);
}

__global__ void k_gcn_self(const float* __restrict__ hw, const float* dis, float* out) {
  long i = (long)blockIdx.x * blockDim.x + threadIdx.x;
  if (i < (long)NN * GCN_H) {
    int n = (int)(i / GCN_H);
    float w = dis[n] * dis[n];
    out[i] += hw[i] * w;
  }
}

__global__ void k_att_scores(const float* __restrict__ hg,
                             const float* __restrict__ att_s,
                             const float* __restrict__ att_d,
                             float* as_, float* ad_) {
  int i = blockIdx.x * blockDim.x + threadIdx.x;
  if (i >= NN * HEADS) return;
  int n = i / HEADS, h = i % HEADS;
  const float* row = hg + (long)n * HID + h * GAT_H;
  const float* ws = att_s + h * GAT_H;
  const float* wd = att_d + h * GAT_H;
  float s = 0.f, dv = 0.f;
  for (int c = 0; c < GAT_H; ++c) { float v = row[c]; s += v * ws[c]; dv += v * wd[c]; }
  as_[i] = s; ad_[i] = dv;
}

__global__ void k_att_max(const float* as_, const float* ad_, const int* s,
                          const int* d, int E, int E2, unsigned* emaxu, float* evbuf) {
  int i = blockIdx.x * blockDim.x + threadIdx.x;
  if (i >= E2 * HEADS) return;
  int e = i / HEADS, h = i % HEADS;
  int si, di;
  if (e < E) { si = s[e]; di = d[e]; } else { si = di = e - E; }
  float ev = as_[si * HEADS + h] + ad_[di * HEADS + h];
  ev = ev > 0.f ? ev : ev * NEG_SLOPE;
  evbuf[i] = ev;
  int b = __float_as_int(ev);
  unsigned enc = (unsigned)b ^ ((b < 0) ? 0xFFFFFFFFu : 0x80000000u);
  atomicMax(&emaxu[di * HEADS + h], enc);
}

__global__ void k_decode_max(const unsigned* emaxu, float* emaxf, int n) {
  int i = blockIdx.x * blockDim.x + threadIdx.x;
  if (i < n) {
    unsigned u = emaxu[i];
    unsigned b = (u & 0x80000000u) ? (u ^ 0x80000000u) : ~u;
    emaxf[i] = __int_as_float((int)b);
  }
}

__global__ void k_att_expsum(const int* s, const int* d, int E, int E2,
                             const float* emaxf, float* evbuf, float* denom) {
  int i = blockIdx.x * blockDim.x + threadIdx.x;
  if (i >= E2 * HEADS) return;
  int e = i / HEADS, h = i % HEADS;
  int di = (e < E) ? d[e] : (e - E);
  float ex = expf(evbuf[i] - emaxf[di * HEADS + h]);
  evbuf[i] = ex;
  atomicAdd(&denom[di * HEADS + h], ex);
}

__global__ void k_gat_scatter(const float* __restrict__ hg, const int* s,
                              const int* d, int E, const float* evbuf,
                              const float* denom, float* out) {
  __shared__ float al[HEADS];
  __shared__ int sd[2];
  int e = blockIdx.x;
  if (threadIdx.x < 2) {
    int v;
    if (e < E) v = (threadIdx.x == 0) ? s[e] : d[e];
    else v = e - E;
    sd[threadIdx.x] = v;
  }
  __syncthreads();
  int si = sd[0], di = sd[1];
  if (threadIdx.x < HEADS)
    al[threadIdx.x] = evbuf[e * HEADS + threadIdx.x] / denom[di * HEADS + threadIdx.x];
  __syncthreads();
  const float* src = hg + (long)si * HID;
  float* dst = out + (long)di * HID;
  for (int idx = threadIdx.x; idx < HID; idx += blockDim.x)
    atomicAdd(&dst[idx], src[idx] * al[idx >> 8]);  // GAT_H=256 -> head = idx/256
}

__global__ void k_pool(const float* __restrict__ hF, const int* batch,
                       float* pooled, float* counts) {
  long i = (long)blockIdx.x * blockDim.x + threadIdx.x;
  if (i >= (long)NN * HID) return;
  int n = (int)(i / HID); int f = (int)(i % HID);
  int g = batch[n];
  atomicAdd(&pooled[(long)g * HID + f], hF[i]);
  if (f == 0) atomicAdd(&counts[g], 1.0f);
}

__global__ void k_final(const float* pooled, const float* counts,
                        const float* W, const float* b, float* out) {
  int i = blockIdx.x * blockDim.x + threadIdx.x;
  if (i >= NG * OUT_DIM) return;
  int g = i / OUT_DIM, o = i % OUT_DIM;
  float inv = 1.f / fmaxf(counts[g], 1.f);
  float s = 0.f;
  for (int f = 0; f < HID; ++f) s += pooled[g * HID + f] * W[f * OUT_DIM + o];
  out[i] = s * inv + b[o];
}

// ---------------- host orchestration ----------------
extern "C" void kernel_launch(void* const* d_in, const int* in_sizes, int n_in,
                              void* d_out, int out_size, void* d_ws, size_t ws_size,
                              hipStream_t stream) {
  (void)n_in; (void)out_size; (void)ws_size;
  const float* x      = (const float*)d_in[0];
  const int*   eidx   = (const int*)d_in[1];
  const int*   batch  = (const int*)d_in[2];
  const float* W_gcn  = (const float*)d_in[3];
  const float* b_gcn  = (const float*)d_in[4];
  const float* W_gat  = (const float*)d_in[5];
  const float* att_s  = (const float*)d_in[6];
  const float* att_d  = (const float*)d_in[7];
  const float* b_gat  = (const float*)d_in[8];
  const float* W_out  = (const float*)d_in[9];
  const float* b_out  = (const float*)d_in[10];
  const int E  = in_sizes[1] / 2;
  const int E2 = E + NN;
  const int* s_arr = eidx;
  const int* d_arr = eidx + E;

  char* w = (char*)d_ws;
  auto carve = [&](size_t bytes) -> char* {
    char* p = w; w += (bytes + 255) & ~(size_t)255; return p;
  };
  float*    A      = (float*)carve((size_t)NN * NN * 4);
  float*    V      = (float*)carve((size_t)NN * NN * 4);
  float*    U      = (float*)carve((size_t)NN * NN * 4);
  float*    Jws    = (float*)carve((size_t)NPAIR * PW * PW * 4);
  float*    degL   = (float*)carve(NN * 4);
  float*    key    = (float*)carve(NN * 4);
  int*      sidx   = (int*)carve(NN * 4);
  float*    ht     = (float*)carve((size_t)NN * F_INP * 4);
  float*    hw     = (float*)carve((size_t)NN * GCN_H * 4);
  float*    hgcn   = (float*)carve((size_t)NN * GCN_H * 4);
  float*    hg     = (float*)carve((size_t)NN * HID * 4);
  float*    hgat   = (float*)carve((size_t)NN * HID * 4);
  float*    hF     = (float*)carve((size_t)NN * HID * 4);
  float*    dis    = (float*)carve(NN * 4);
  float*    as_    = (float*)carve((size_t)NN * HEADS * 4);
  float*    ad_    = (float*)carve((size_t)NN * HEADS * 4);
  unsigned* emaxu  = (unsigned*)carve((size_t)NN * HEADS * 4);
  float*    emaxf  = (float*)carve((size_t)NN * HEADS * 4);
  float*    denom  = (float*)carve((size_t)NN * HEADS * 4);
  float*    evbuf  = (float*)carve((size_t)E2 * HEADS * 4);
  float*    pooled = (float*)carve((size_t)NG * HID * 4);
  float*    counts = (float*)carve(NG * 4);

  const size_t K1S = (size_t)(2 * 128 * GS + 128 * TS2 + 128) * 4 + 128 * 4;
  const size_t K2S = (size_t)(128 * JS + 64 * CS) * 4;
  hipFuncSetAttribute((const void*)k_gram_jacobi,
                      hipFuncAttributeMaxDynamicSharedMemorySize, (int)K1S);
  hipFuncSetAttribute((const void*)k_apply_rot,
                      hipFuncAttributeMaxDynamicSharedMemorySize, (int)K2S);

  const long NN2 = (long)NN * NN;
  dim3 blk(256);

  // ---- Laplacian ----
  k_fill<<<(int)((NN2 + 255) / 256), blk, 0, stream>>>(A, 0.f, NN2);
  k_set_adj<<<(E + 255) / 256, blk, 0, stream>>>(A, s_arr, d_arr, E);
  k_row_sum<<<NN, blk, 0, stream>>>(A, degL);
  k_form_lap<<<(int)((NN2 + 255) / 256), blk, 0, stream>>>(A, degL);

  // ---- eigendecomposition: block one-sided Jacobi ----
  k_fill<<<(int)((NN2 + 255) / 256), blk, 0, stream>>>(V, 0.f, NN2);
  k_set_diag<<<(NN + 255) / 256, blk, 0, stream>>>(V);
  for (int sw = 0; sw < OUTER_SWEEPS; ++sw) {
    for (int r = 0; r < NROUND; ++r) {
      k_gram_jacobi<<<NPAIR, blk, K1S, stream>>>(A, Jws, r);
      k_apply_rot<<<dim3(NPAIR, NN / 64), blk, K2S, stream>>>(A, V, Jws, r);
    }
  }

  // ---- sort eigenpairs ascending, permute into U ----
  k_col_norm<<<NN, blk, 0, stream>>>(A, key);
  k_iota<<<(NN + 255) / 256, blk, 0, stream>>>(sidx);
  for (int k = 2; k <= NN; k <<= 1)
    for (int j = k >> 1; j > 0; j >>= 1)
      k_bitonic<<<NN / 256, blk, 0, stream>>>(key, sidx, j, k);
  k_permute_cols<<<(int)((NN2 + 255) / 256), blk, 0, stream>>>(V, U, sidx);

  // ---- GFT: ht = U^T x ----
  k_gemm<<<dim3(F_INP / 64, NN / 64), blk, 0, stream>>>(U, x, ht, NN, F_INP, NN, NN, 1);

  // ---- GCN: hw = ht @ W_gcn, symmetric-norm aggregate, bias+relu ----
  k_gemm<<<dim3(GCN_H / 64, NN / 64), blk, 0, stream>>>(ht, W_gcn, hw, NN, GCN_H,
                                                        F_INP, F_INP, 0);
  k_fill<<<(NN + 255) / 256, blk, 0, stream>>>(dis, 1.f, NN);
  k_edge_deg<<<(E + 255) / 256, blk, 0, stream>>>(d_arr, E, dis);
  k_rsqrt<<<(NN + 255) / 256, blk, 0, stream>>>(dis, NN);
  k_fill<<<(int)(((long)NN * GCN_H + 255) / 256), blk, 0, stream>>>(hgcn, 0.f,
                                                                    (long)NN * GCN_H);
  k_gcn_scatter<<<E, blk, 0, stream>>>(hw, s_arr, d_arr, dis, hgcn, E);
  k_gcn_self<<<(int)(((long)NN * GCN_H + 255) / 256), blk, 0, stream>>>(hw, dis, hgcn);
  k_bias_relu<<<(int)(((long)NN * GCN_H + 255) / 256), blk, 0, stream>>>(
      hgcn, b_gcn, (long)NN * GCN_H, GCN_H);

  // ---- GAT ----
  k_gemm<<<dim3(HID / 64, NN / 64), blk, 0, stream>>>(hgcn, W_gat, hg, NN, HID,
                                                      GCN_H, GCN_H, 0);
  k_att_scores<<<(NN * HEADS + 255) / 256, blk, 0, stream>>>(hg, att_s, att_d, as_, ad_);
  k_fill<<<(NN * HEADS + 255) / 256, blk, 0, stream>>>((float*)emaxu, 0.f, NN * HEADS);
  k_att_max<<<(E2 * HEADS + 255) / 256, blk, 0, stream>>>(as_, ad_, s_arr, d_arr, E,
                                                          E2, emaxu, evbuf);
  k_decode_max<<<(NN * HEADS + 255) / 256, blk, 0, stream>>>(emaxu, emaxf, NN * HEADS);
  k_fill<<<(NN * HEADS + 255) / 256, blk, 0, stream>>>(denom, 0.f, NN * HEADS);
  k_att_expsum<<<(E2 * HEADS + 255) / 256, blk, 0, stream>>>(s_arr, d_arr, E, E2,
                                                             emaxf, evbuf, denom);
  k_fill<<<(int)(((long)NN * HID + 255) / 256), blk, 0, stream>>>(hgat, 0.f,
                                                                  (long)NN * HID);
  k_gat_scatter<<<E2, blk, 0, stream>>>(hg, s_arr, d_arr, E, evbuf, denom, hgat);
  k_bias_relu<<<(int)(((long)NN * HID + 255) / 256), blk, 0, stream>>>(
      hgat, b_gat, (long)NN * HID, HID);

  // ---- iGFT: hF = U @ hgat ----
  k_gemm<<<dim3(HID / 64, NN / 64), blk, 0, stream>>>(U, hgat, hF, NN, HID, NN, NN, 0);

  // ---- pool + output linear ----
  k_fill<<<(int)(((long)NG * HID + 255) / 256), blk, 0, stream>>>(pooled, 0.f,
                                                                  (long)NG * HID);
  k_fill<<<1, blk, 0, stream>>>(counts, 0.f, NG);
  k_pool<<<(int)(((long)NN * HID + 255) / 256), blk, 0, stream>>>(hF, batch, pooled,
                                                                  counts);
  k_final<<<(NG * OUT_DIM + 255) / 256, blk, 0, stream>>>(pooled, counts, W_out,
                                                          b_out, (float*)d_out);
}